// MHARot_78743930405212
// MI455X (gfx1250) — compile-verified
//
#include <hip/hip_runtime.h>
#include <hip/hip_bf16.h>
#include <math.h>

// Problem constants (from reference)
constexpr int Bc = 2, Tc = 2048, Ec = 1024, Hc = 16, Dc = 64;
constexpr int M_TOT = Bc * Tc;     // 4096 rows of x
constexpr int N_TOT = 3 * Ec;      // 3072 output cols of qkv
constexpr int NSUB  = 4;           // 4 x 16 = 64 cols per wave in GEMM
constexpr size_t QKV_ELEMS = (size_t)Bc * Hc * Tc * Dc;  // 4M floats each

typedef __attribute__((ext_vector_type(2))) float v2f;
typedef __attribute__((ext_vector_type(8))) float v8f;
typedef __attribute__((ext_vector_type(4))) int v4i;

__device__ __forceinline__ v8f wmma_f32_16x16x4(v2f a, v2f b, v8f c) {
  // 8 args: (neg_a, A, neg_b, B, c_mod, C, reuse_a, reuse_b)
  return __builtin_amdgcn_wmma_f32_16x16x4_f32(false, a, false, b, (short)0, c,
                                               false, false);
}

// ---- CDNA5 async global->LDS path (guarded; falls back to sync staging) ----
#if defined(__has_builtin)
#if __has_builtin(__builtin_amdgcn_global_load_async_to_lds_b128)
#define HAVE_ASYNC 1
#endif
#endif
#ifndef HAVE_ASYNC
#define HAVE_ASYNC 0
#endif

#if HAVE_ASYNC
#if __has_builtin(__builtin_amdgcn_s_wait_asynccnt)
#define WAIT_ASYNC(n)                                                         \
  do {                                                                        \
    __builtin_amdgcn_s_wait_asynccnt(n);                                      \
    asm volatile("" ::: "memory");                                            \
  } while (0)
#else
#define WAIT_ASYNC(n) asm volatile("s_wait_asynccnt %0" ::"i"(n) : "memory")
#endif
// Builtin signature (from compiler diagnostic): param 0 is `int4 __device__*`
// (global/AS1 pointer to a 16-byte vector). LDS side is AS3 int4*.
typedef __attribute__((address_space(1))) v4i* gv4_ptr;
typedef __attribute__((address_space(3))) v4i* lv4_ptr;
#else
#define WAIT_ASYNC(n) asm volatile("" ::: "memory")
#endif

// Copy 4KB (1024 contiguous floats) global -> LDS, one wave, 16B per lane.
__device__ __forceinline__ void stage_tile_4kb(const float* __restrict__ g,
                                               float* l, int lane) {
#if HAVE_ASYNC
  gv4_ptr gp = (gv4_ptr)g;
  lv4_ptr lp = (lv4_ptr)l;
#pragma unroll
  for (int i = 0; i < 8; ++i) {
    const int c = i * 32 + lane;  // 16-byte chunk index, 256 chunks total
    __builtin_amdgcn_global_load_async_to_lds_b128(gp + c, lp + c, 0, 0);
  }
#else
#pragma unroll
  for (int i = 0; i < 8; ++i) {
    const int c = (i * 32 + lane) * 4;
    *(float4*)(l + c) = *(const float4*)(g + c);
  }
#endif
}

// ---------------------------------------------------------------------------
// Flash attention. One wave per (b, h, 16-row q tile). K/V tiles are
// double-buffered in LDS via async global->LDS DMA (ASYNCcnt); tile kt+1's
// copy is issued before computing tile kt. All matrix math is f32 WMMA.
// __launch_bounds__(256, 1): let the allocator use the ~200 VGPRs this
// kernel needs (wave32 allows 1024) instead of spilling qa[] to scratch.
// ---------------------------------------------------------------------------
__global__ __launch_bounds__(256, 1) void attn_kernel(
    const float* __restrict__ Q, const float* __restrict__ Km,
    const float* __restrict__ V, float* __restrict__ out) {
  const int lane  = threadIdx.x & 31;
  const int wib   = threadIdx.x >> 5;
  const int waves = blockDim.x >> 5;
  const int gw    = blockIdx.x * waves + wib;

  const int qTiles = Tc / 16;  // 128
  const int bh     = gw / qTiles;
  const int qt     = gw % qTiles;
  if (bh >= Bc * Hc) return;
  const int b = bh / Hc;
  const int h = bh % Hc;

  const float* Qbh = Q + (size_t)bh * Tc * Dc;
  const float* Kbh = Km + (size_t)bh * Tc * Dc;
  const float* Vbh = V + (size_t)bh * Tc * Dc;

  const int mrow  = qt * 16 + (lane & 15);
  const int khalf = (lane >> 4) * 2;
  const int ncol  = lane & 15;

  // Q tile in WMMA-A layout: 16 K-steps of float2 per lane (covers D=64)
  v2f qa[16];
#pragma unroll
  for (int j = 0; j < 16; ++j)
    qa[j] = *(const v2f*)(Qbh + (size_t)mrow * Dc + 4 * j + khalf);

  v8f o[4];
#pragma unroll
  for (int nt = 0; nt < 4; ++nt) o[nt] = (v8f){0, 0, 0, 0, 0, 0, 0, 0};
  float mstat[8], lstat[8];
#pragma unroll
  for (int v = 0; v < 8; ++v) { mstat[v] = -INFINITY; lstat[v] = 0.0f; }

  // Per-wave LDS: double-buffered K/V tiles + P relayout scratch.
  __shared__ float ktile[8][2][16 * 64];
  __shared__ float vtile[8][2][16 * 64];
  __shared__ float pbuf[8][16 * 16];
  float* pl = &pbuf[wib][0];

  const float scale = 0.125f;  // 1/sqrt(64)
  const int nkt = Tc / 16;     // 128

  // One k-tile step: S = Q K^T, online softmax, O = O*alpha + P V.
  auto tile_step = [&](const float* kb, const float* vb) {
    // ---- S = Q K^T : two independent WMMA chains, B operands from LDS ----
    const float* kbrow = kb + ncol * 64 + khalf;
    v8f s0 = (v8f){0, 0, 0, 0, 0, 0, 0, 0};
    v8f s1 = (v8f){0, 0, 0, 0, 0, 0, 0, 0};
#pragma unroll
    for (int j = 0; j < 8; ++j) {
      v2f b0 = *(const v2f*)(kbrow + 4 * (2 * j));
      v2f b1 = *(const v2f*)(kbrow + 4 * (2 * j + 1));
      s0 = wmma_f32_16x16x4(qa[2 * j], b0, s0);
      s1 = wmma_f32_16x16x4(qa[2 * j + 1], b1, s1);
    }
    const v8f s = s0 + s1;

    // ---- online softmax update (row stats via 16-lane butterflies) ----
    float pcols[8], alpha[8];
#pragma unroll
    for (int v = 0; v < 8; ++v) {
      float sv = s[v] * scale;
      float rm = sv;
      rm = fmaxf(rm, __shfl_xor(rm, 1, 32));
      rm = fmaxf(rm, __shfl_xor(rm, 2, 32));
      rm = fmaxf(rm, __shfl_xor(rm, 4, 32));
      rm = fmaxf(rm, __shfl_xor(rm, 8, 32));
      const float mnew = fmaxf(mstat[v], rm);
      const float p    = __expf(sv - mnew);
      float rs = p;
      rs += __shfl_xor(rs, 1, 32);
      rs += __shfl_xor(rs, 2, 32);
      rs += __shfl_xor(rs, 4, 32);
      rs += __shfl_xor(rs, 8, 32);
      const float a = __expf(mstat[v] - mnew);
      lstat[v] = lstat[v] * a + rs;
      mstat[v] = mnew;
      alpha[v] = a;
      pcols[v] = p;
    }

    // ---- relayout P (C layout) -> A layout through LDS ----
    {
      const int mbase = 8 * (lane >> 4);
#pragma unroll
      for (int v = 0; v < 8; ++v) pl[(mbase + v) * 16 + ncol] = pcols[v];
    }
    asm volatile("s_wait_dscnt 0x0" ::: "memory");
    v2f pa[4];
#pragma unroll
    for (int j = 0; j < 4; ++j) {
      const int r = (lane & 15) * 16 + 4 * j + khalf;
      pa[j].x = pl[r];
      pa[j].y = pl[r + 1];
    }

    // ---- O = O*alpha + P @ V (V read from LDS) ----
#pragma unroll
    for (int nt = 0; nt < 4; ++nt) {
      v2f vbop[4];
#pragma unroll
      for (int j = 0; j < 4; ++j) {
        const int krow = 4 * j + khalf;
        vbop[j].x = vb[krow * 64 + nt * 16 + ncol];
        vbop[j].y = vb[(krow + 1) * 64 + nt * 16 + ncol];
      }
      v8f c;
#pragma unroll
      for (int v = 0; v < 8; ++v) c[v] = o[nt][v] * alpha[v];
#pragma unroll
      for (int j = 0; j < 4; ++j) c = wmma_f32_16x16x4(pa[j], vbop[j], c);
      o[nt] = c;
    }
  };

  // Prologue: start DMA of tile 0.
  stage_tile_4kb(Kbh, &ktile[wib][0][0], lane);
  stage_tile_4kb(Vbh, &vtile[wib][0][0], lane);

  // Steady state: stage kt+1, wait for kt, compute kt. No in-loop branch.
  for (int kt = 0; kt < nkt - 1; ++kt) {
    const int cur = kt & 1;
    stage_tile_4kb(Kbh + (size_t)(kt + 1) * 16 * Dc, &ktile[wib][cur ^ 1][0],
                   lane);
    stage_tile_4kb(Vbh + (size_t)(kt + 1) * 16 * Dc, &vtile[wib][cur ^ 1][0],
                   lane);
    WAIT_ASYNC(16);  // 16 just issued may remain; tile kt is complete
    tile_step(&ktile[wib][cur][0], &vtile[wib][cur][0]);
  }
  // Peeled last tile.
  WAIT_ASYNC(0);
  tile_step(&ktile[wib][(nkt - 1) & 1][0], &vtile[wib][(nkt - 1) & 1][0]);

  // ---- finalize: out[b][t][h*D + d] = O / l ----
#pragma unroll
  for (int nt = 0; nt < 4; ++nt) {
#pragma unroll
    for (int v = 0; v < 8; ++v) {
      const int q = qt * 16 + v + 8 * (lane >> 4);
      out[(((size_t)b * Tc + q) * Hc + h) * Dc + nt * 16 + ncol] =
          o[nt][v] / lstat[v];
    }
  }
}

// ---------------------------------------------------------------------------
// QKV GEMM: qkv = x @ W^T + b, scattered into Q/K/V as [B,H,T,D].
// One wave computes 16(M) x 64(N); operand loads run through a PF-deep
// register ring: steady-state loop is branch-free, last PF stages drain.
// ---------------------------------------------------------------------------
__global__ void qkv_gemm_kernel(const float* __restrict__ x,
                                const float* __restrict__ W,
                                const float* __restrict__ bias,
                                float* __restrict__ Q,
                                float* __restrict__ K,
                                float* __restrict__ V) {
  const int lane  = threadIdx.x & 31;
  const int wib   = threadIdx.x >> 5;
  const int waves = blockDim.x >> 5;
  const int gw    = blockIdx.x * waves + wib;

  const int nMacro    = N_TOT / (16 * NSUB);  // 48
  const int mTile     = gw / nMacro;          // 0..255
  const int nMacroIdx = gw % nMacro;
  if (mTile >= M_TOT / 16) return;

  const int m0 = mTile * 16;
  const int n0 = nMacroIdx * 16 * NSUB;

  const int mrow  = m0 + (lane & 15);  // A-operand row for this lane
  const int khalf = (lane >> 4) * 2;   // K sub-slot (0 or 2)
  const int ncol  = lane & 15;         // B/C-operand column for this lane

  const float* xrow = x + (size_t)mrow * Ec + khalf;
  const float* wrow[NSUB];
#pragma unroll
  for (int s = 0; s < NSUB; ++s)
    wrow[s] = W + (size_t)(n0 + s * 16 + ncol) * Ec + khalf;

  v8f acc[NSUB];
#pragma unroll
  for (int s = 0; s < NSUB; ++s) acc[s] = (v8f){0, 0, 0, 0, 0, 0, 0, 0};

  // Prefetch ring: PF k-steps in flight.
  constexpr int PF = 4;
  v2f a_pf[PF];
  v2f b_pf[PF][NSUB];
#pragma unroll
  for (int p = 0; p < PF; ++p) {
    a_pf[p] = *(const v2f*)(xrow + 4 * p);
#pragma unroll
    for (int s = 0; s < NSUB; ++s) b_pf[p][s] = *(const v2f*)(wrow[s] + 4 * p);
  }

  // Steady state: consume stage p, then refill it PF steps ahead.
  for (int k = 0; k < Ec - 4 * PF; k += 4 * PF) {
#pragma unroll
    for (int p = 0; p < PF; ++p) {
#pragma unroll
      for (int s = 0; s < NSUB; ++s)
        acc[s] = wmma_f32_16x16x4(a_pf[p], b_pf[p][s], acc[s]);
      const int kn = k + 4 * (p + PF);
      a_pf[p] = *(const v2f*)(xrow + kn);
#pragma unroll
      for (int s = 0; s < NSUB; ++s) b_pf[p][s] = *(const v2f*)(wrow[s] + kn);
    }
  }
  // Drain the ring (no more loads).
#pragma unroll
  for (int p = 0; p < PF; ++p)
#pragma unroll
    for (int s = 0; s < NSUB; ++s)
      acc[s] = wmma_f32_16x16x4(a_pf[p], b_pf[p][s], acc[s]);

  // Epilogue: add bias, scatter into Q/K/V[b][h][t][d]
#pragma unroll
  for (int s = 0; s < NSUB; ++s) {
    const int n     = n0 + s * 16 + ncol;  // global qkv column
    const float bv  = bias[n];
    const int which = n >> 10;             // 0=q 1=k 2=v (uniform per wave)
    const int h     = (n & 1023) >> 6;     // head (uniform per wave)
    const int d     = n & 63;
    float* base = (which == 0) ? Q : (which == 1 ? K : V);
#pragma unroll
    for (int v = 0; v < 8; ++v) {
      const int m  = m0 + v + 8 * (lane >> 4);  // row held by this VGPR
      const int bb = m >> 11;                   // m / T
      const int t  = m & (Tc - 1);
      base[(((size_t)(bb * Hc + h) * Tc + t) * Dc) + d] = acc[s][v] + bv;
    }
  }
}

// ---------------------------------------------------------------------------
// RoPE in place on Q and K ([B,H,T,D], pairs (2i, 2i+1) along D)
// ---------------------------------------------------------------------------
__global__ void rope_kernel(float* __restrict__ Q, float* __restrict__ K) {
  const size_t npairs = (size_t)Bc * Hc * Tc * (Dc / 2);
  size_t idx = (size_t)blockIdx.x * blockDim.x + threadIdx.x;
  if (idx >= 2 * npairs) return;
  float* ptr = (idx < npairs) ? Q : K;
  size_t p   = (idx < npairs) ? idx : idx - npairs;

  const int    i    = (int)(p & (Dc / 2 - 1));  // pair index 0..31
  const size_t rest = p >> 5;                   // [b][h][t] flat, t fastest
  const int    t    = (int)(rest & (Tc - 1));

  const float freq = __expf(-((float)(2 * i) / (float)Dc) * __logf(10000.0f));
  float s, c;
  __sincosf((float)t * freq, &s, &c);

  float* base = ptr + rest * Dc + 2 * i;
  const float xr = base[0];
  const float xi = base[1];
  base[0] = xr * c - xi * s;
  base[1] = xr * s + xi * c;
}

// ---------------------------------------------------------------------------
extern "C" void kernel_launch(void* const* d_in, const int* in_sizes, int n_in,
                              void* d_out, int out_size, void* d_ws,
                              size_t ws_size, hipStream_t stream) {
  (void)in_sizes; (void)n_in; (void)out_size; (void)ws_size;
  const float* x    = (const float*)d_in[0];
  const float* W    = (const float*)d_in[1];
  const float* bias = (const float*)d_in[2];
  float* ws  = (float*)d_ws;
  float* Q   = ws;
  float* K   = ws + QKV_ELEMS;
  float* V   = ws + 2 * QKV_ELEMS;
  float* out = (float*)d_out;

  // Kernel 1: 256 M-tiles x 48 N-macro-tiles = 12288 waves, 8 waves/block
  {
    const int totalWaves = (M_TOT / 16) * (N_TOT / (16 * NSUB));
    qkv_gemm_kernel<<<totalWaves / 8, 256, 0, stream>>>(x, W, bias, Q, K, V);
  }
  // Kernel 2: one thread per rotation pair, on Q and K
  {
    const size_t threads = 2 * (size_t)Bc * Hc * Tc * (Dc / 2);
    rope_kernel<<<(unsigned)((threads + 255) / 256), 256, 0, stream>>>(Q, K);
  }
  // Kernel 3: one wave per (b, h, q-tile) = 4096 waves, 8 waves/block
  {
    const int totalWaves = Bc * Hc * (Tc / 16);
    attn_kernel<<<totalWaves / 8, 256, 0, stream>>>(Q, K, V, out);
  }
}